// FastRCNNOutputLayers_34325378629777
// MI455X (gfx1250) — compile-verified
//
#include <hip/hip_runtime.h>
#include <hip/hip_bf16.h>
#include <stdint.h>

#define RR 1000
#define KK 80
#define NPAD 1024
#define TILE 256          // mask tile rows
#define TOPK_N 100
#define NCAND (KK * TOPK_N)   // 8000 compacted candidate slots
#define IMG_W 1333.0f
#define IMG_H 800.0f
#define NMS_TH 0.5f
#define SCORE_TH 0.05f
#define NCOLS 87          // 4 box + 1 score + 1 cls + 81 full scores
#define SROW  (KK + 1)    // 81
#define BROW  (KK * 4)    // 320

typedef unsigned int       u32;
typedef unsigned long long u64;
typedef u32 v4u __attribute__((ext_vector_type(4)));
typedef int v4i __attribute__((ext_vector_type(4)));
typedef int v8i __attribute__((ext_vector_type(8)));

__device__ __forceinline__ u32 f2ord(float f) {
    u32 u = __float_as_uint(f);
    return (u & 0x80000000u) ? ~u : (u | 0x80000000u);
}
__device__ __forceinline__ float ord2f(u32 o) {
    u32 u = (o & 0x80000000u) ? (o & 0x7fffffffu) : ~o;
    return __uint_as_float(u);
}

// ---------------------------------------------------------------------------
// Kernel 0: threshold = min(0.05, (mean(fg)+max(fg))/2) over fg = scores[:, :80]
// Division-free layout: wave w owns rows w, w+32, ...; lanes stride columns.
// ---------------------------------------------------------------------------
__global__ void __launch_bounds__(1024)
thres_kernel(const float* __restrict__ scores, float* __restrict__ thres_out) {
    __shared__ float ssum[1024];
    __shared__ float smax[1024];
    const int tid  = threadIdx.x;
    const int lane = tid & 31;
    const int wv32 = tid >> 5;
    float acc = 0.0f, mx = -INFINITY;
    for (int r = wv32; r < RR; r += 32) {
        const float* row = scores + (size_t)r * SROW;
        for (int c0 = lane; c0 < KK; c0 += 32) {    // skips background col 80
            float v = row[c0];
            acc += v;
            mx = fmaxf(mx, v);
        }
    }
    ssum[tid] = acc; smax[tid] = mx;
    __syncthreads();
    for (int off = 512; off > 0; off >>= 1) {
        if (tid < off) {
            ssum[tid] += ssum[tid + off];
            smax[tid] = fmaxf(smax[tid], smax[tid + off]);
        }
        __syncthreads();
    }
    if (tid == 0) {
        float mean = ssum[0] / (float)(RR * KK);
        thres_out[0] = fminf(SCORE_TH, 0.5f * (mean + smax[0]));
    }
}

// ---------------------------------------------------------------------------
// Kernel 1: per-class NMS (one block = one class = one WGP).
//   - TDM stages the strided box column (tile 4x1000, stride 320 elems) to LDS
//   - u64-key bitonic sort (descending score, stable index)
//   - ballot-built suppression bitmask tiles + single-wave register scan
//   - wave32 prefix-scan compaction: top-100 survivors per class emitted
// ---------------------------------------------------------------------------
union ShmU {
    struct { float stage[RR * 4]; u64 keys[NPAD]; } a;  // 24,192 B (dead after gather)
    u32 mask[TILE * 32];                                // 32,768 B (tile of mask rows)
};

__global__ void __launch_bounds__(1024)
nms_kernel(const float* __restrict__ boxes, const float* __restrict__ scores,
           const float* __restrict__ thres_p,
           float* __restrict__ cand_val, int* __restrict__ cand_ord) {
    __shared__ ShmU shm;
    __shared__ float sx1[NPAD], sy1[NPAD], sx2[NPAD], sy2[NPAD], sar[NPAD];
    __shared__ u32 validw[32];
    __shared__ u32 wpre[32];

    const int tid  = threadIdx.x;
    const int lane = tid & 31;
    const int wv32 = tid >> 5;
    const int c    = blockIdx.x;

    // ---- stage boxes[:, c*4 .. c*4+3] into LDS via the Tensor Data Mover ----
#if __has_builtin(__builtin_amdgcn_tensor_load_to_lds)
    if (tid < 32) {  // TDM ignores EXEC; issue from one wave only
        u32 lds_off = (u32)(uintptr_t)(&shm.a.stage[0]);
        u64 gaddr   = (u64)(uintptr_t)(boxes + (size_t)c * 4);
        v4u g0 = { 1u,                                   // count=1 valid descriptor
                   lds_off,                              // lds_addr (bytes)
                   (u32)(gaddr & 0xffffffffu),           // global_addr[31:0]
                   (u32)((gaddr >> 32) & 0x01ffffffu) | (2u << 30) }; // addr[56:32] | type=2
        v8i g1 = { (int)(2u << 16),        // data_size=4B, no multicast/pad/iterate
                   (int)(4u    << 16),     // tensor_dim0 = 4 elements (one box)
                   (int)(1000u << 16),     // tensor_dim0 hi=0 | tensor_dim1 lo = 1000
                   (int)(4u    << 16),     // tensor_dim1 hi=0 | tile_dim0 = 4
                   (int)1000,              // tile_dim1 = 1000, tile_dim2 = 0
                   (int)320,               // tensor_dim0_stride = 320 elements
                   0, 0 };
        v4i g2 = {0, 0, 0, 0};
        v4i g3 = {0, 0, 0, 0};
#if __has_include(<hip/amd_detail/amd_gfx1250_TDM.h>)
        v8i g4 = {0, 0, 0, 0, 0, 0, 0, 0};
        __builtin_amdgcn_tensor_load_to_lds(g0, g1, g2, g3, g4, 0);
#else
        __builtin_amdgcn_tensor_load_to_lds(g0, g1, g2, g3, 0);
#endif
        __builtin_amdgcn_s_wait_tensorcnt(0);
    }
#elif defined(__HIP_DEVICE_COMPILE__)
#error "gfx1250 TDM builtin __builtin_amdgcn_tensor_load_to_lds not available (device pass)"
#else
    // Host pass: __global__ bodies are parsed but never executed on host.
    for (int e = tid; e < RR * 4; e += 1024)
        shm.a.stage[e] = boxes[(size_t)(e >> 2) * BROW + c * 4 + (e & 3)];
#endif

    // ---- init compacted candidate slots for this class ----
    if (tid < TOPK_N) {
        cand_val[c * TOPK_N + tid] = -INFINITY;
        cand_ord[c * TOPK_N + tid] = 0;
    }

    // ---- build sort keys (ascending key == descending score, stable index) ----
    const float thres = thres_p[0];
    {
        float masked = -INFINITY;
        if (tid < RR) {
            __builtin_prefetch(&scores[(size_t)tid * SROW + c], 0, 0);
            float fgv = scores[(size_t)tid * SROW + c];
            if (fgv > thres) masked = fgv;
        }
        shm.a.keys[tid] = ((u64)(~f2ord(masked)) << 32) | (u32)tid;
    }
    __syncthreads();

    // ---- bitonic sort, ascending keys (1024 elements, 1 per thread) ----
    for (int k = 2; k <= NPAD; k <<= 1) {
        for (int j = k >> 1; j > 0; j >>= 1) {
            int ixj = tid ^ j;
            if (ixj > tid) {
                bool up = ((tid & k) == 0);
                u64 a = shm.a.keys[tid], b = shm.a.keys[ixj];
                if ((a > b) == up) { shm.a.keys[tid] = b; shm.a.keys[ixj] = a; }
            }
            __syncthreads();
        }
    }

    // ---- decode, gather clipped boxes into sorted order ----
    u64 kk = shm.a.keys[tid];
    u32 oidx = (u32)kk;                       // original proposal index (<1000 for tid<1000)
    float sc = ord2f(~(u32)(kk >> 32));       // sorted (masked) score
    bool valid = (sc != -INFINITY);
    {
        float bx1 = fminf(fmaxf(shm.a.stage[oidx * 4 + 0], 0.0f), IMG_W);
        float by1 = fminf(fmaxf(shm.a.stage[oidx * 4 + 1], 0.0f), IMG_H);
        float bx2 = fminf(fmaxf(shm.a.stage[oidx * 4 + 2], 0.0f), IMG_W);
        float by2 = fminf(fmaxf(shm.a.stage[oidx * 4 + 3], 0.0f), IMG_H);
        sx1[tid] = bx1; sy1[tid] = by1; sx2[tid] = bx2; sy2[tid] = by2;
        sar[tid] = fmaxf(bx2 - bx1, 0.0f) * fmaxf(by2 - by1, 0.0f);
    }
    if (tid < 32) validw[tid] = 0;
    __syncthreads();          // last read of shm.a before shm.mask overwrites it
    if (valid) atomicOr(&validw[tid >> 5], 1u << (tid & 31));
    __syncthreads();

    // ---- tiled suppression bitmask + single-wave register scan ----
    u32 kw = (tid < 32) ? validw[tid] : 0u;   // wave 0: lane l owns keep bits [l*32, l*32+31]

    for (int t = 0; t < (RR + TILE - 1) / TILE; ++t) {
        const int base = t * TILE;
        const int rows = (RR - base < TILE) ? (RR - base) : TILE;

        // mask build: wave w owns rows base+w, base+w+32, ... ; lanes cover j
        for (int r = wv32; r < rows; r += 32) {
            const int i = base + r;
            float xi1 = sx1[i], yi1 = sy1[i], xi2 = sx2[i], yi2 = sy2[i], ai = sar[i];
            for (int b = 0; b < 32; ++b) {
                int j = b * 32 + lane;        // lane-consecutive -> conflict-free LDS
                bool bit = false;
                if (j > i) {
                    float xx1 = fmaxf(xi1, sx1[j]);
                    float yy1 = fmaxf(yi1, sy1[j]);
                    float xx2 = fminf(xi2, sx2[j]);
                    float yy2 = fminf(yi2, sy2[j]);
                    float inter = fmaxf(xx2 - xx1, 0.0f) * fmaxf(yy2 - yy1, 0.0f);
                    float uni   = ai + sar[j] - inter;
                    bit = (inter / fmaxf(uni, 1e-9f)) > NMS_TH;
                }
                u32 bal = (u32)__ballot(bit); // wave32: low 32 bits
                if (lane == 0) shm.mask[r * 32 + b] = bal;
            }
        }
        __syncthreads();

        // barrier-free serial scan by wave 0; keep mask lives in registers
        if (tid < 32) {
            for (int r = 0; r < rows; ++r) {
                const int i = base + r;
                u32 mrow = shm.mask[r * 32 + lane];
                u32 wvv  = (u32)__shfl((int)kw, i >> 5, 32);
                if ((wvv >> (i & 31)) & 1u) kw &= ~mrow;
            }
        }
        __syncthreads();
    }

    // ---- wave32 prefix-scan of keep popcounts; compact top-100 survivors ----
    if (tid < 32) {
        validw[tid] = kw;
        u32 pc = (u32)__popc(kw);
        u32 scan = pc;
        for (int d = 1; d < 32; d <<= 1) {
            u32 n = (u32)__shfl_up((int)scan, d, 32);
            if (lane >= d) scan += n;
        }
        wpre[tid] = scan - pc;                // exclusive word prefix
    }
    __syncthreads();

    if (tid < RR) {
        u32 w = validw[tid >> 5];
        bool keep = (w >> (tid & 31)) & 1u;
        if (keep) {
            u32 rank = wpre[tid >> 5] + (u32)__popc(w & ((1u << (tid & 31)) - 1u));
            if (rank < TOPK_N) {              // only top-100 per class can reach global top-100
                cand_val[c * TOPK_N + rank] = sc;
                cand_ord[c * TOPK_N + rank] = (int)oidx;
            }
        }
    }
}

// ---------------------------------------------------------------------------
// Kernel 2: global top-100 over 8000 compacted slots + output gather
// ---------------------------------------------------------------------------
__global__ void __launch_bounds__(1024)
topk_kernel(const float* __restrict__ boxes, const float* __restrict__ scores,
            float* __restrict__ cand_val, const int* __restrict__ cand_ord,
            float* __restrict__ out) {
    __shared__ u64 rkey[1024];
    __shared__ u32 sel_flat;
    __shared__ float sel_val;
    const int tid = threadIdx.x;

    for (int t = 0; t < TOPK_N; ++t) {
        u64 best = 0;
        for (int f = tid; f < NCAND; f += 1024) {
            u64 k = ((u64)f2ord(cand_val[f]) << 32) | (u32)(~(u32)f); // ties -> lowest index
            best = (k > best) ? k : best;
        }
        rkey[tid] = best;
        __syncthreads();
        for (int off = 512; off > 0; off >>= 1) {
            if (tid < off) {
                u64 a = rkey[tid], b = rkey[tid + off];
                rkey[tid] = (b > a) ? b : a;
            }
            __syncthreads();
        }
        if (tid == 0) {
            u64 b = rkey[0];
            sel_flat = ~(u32)b;
            sel_val  = ord2f((u32)(b >> 32));
        }
        __syncthreads();

        u32 flat = sel_flat;
        float v  = sel_val;
        bool ok  = (v != -INFINITY);
        float okf = ok ? 1.0f : 0.0f;
        int cls  = (int)(flat / TOPK_N);
        int prop = cand_ord[flat];             // original proposal index in [0,1000)

        if (tid < NCOLS) {
            float o;
            if (tid < 4) {
                float lim = (tid & 1) ? IMG_H : IMG_W;
                o = fminf(fmaxf(boxes[(size_t)prop * BROW + cls * 4 + tid], 0.0f), lim) * okf;
            } else if (tid == 4) {
                o = ok ? v : 0.0f;
            } else if (tid == 5) {
                o = ok ? (float)cls : 0.0f;
            } else {
                o = scores[(size_t)prop * SROW + (tid - 6)] * okf;
            }
            out[t * NCOLS + tid] = o;
        }
        if (tid == 0) cand_val[flat] = -INFINITY;  // kill the selected slot
        __syncthreads();
    }
}

// ---------------------------------------------------------------------------
extern "C" void kernel_launch(void* const* d_in, const int* in_sizes, int n_in,
                              void* d_out, int out_size, void* d_ws, size_t ws_size,
                              hipStream_t stream) {
    (void)in_sizes; (void)n_in; (void)out_size; (void)ws_size;
    const float* boxes  = (const float*)d_in[0];   // (1000, 320)
    const float* scores = (const float*)d_in[1];   // (1000, 81)
    float* out = (float*)d_out;                    // (100, 87)

    float* wsf      = (float*)d_ws;
    float* thres    = wsf;                         // 1 float (16-float slot)
    float* cand_val = wsf + 16;                    // 8000 floats
    int*   cand_ord = (int*)(wsf + 16 + NCAND);    // 8000 ints  (~64 KB total)

    thres_kernel<<<1, 1024, 0, stream>>>(scores, thres);
    nms_kernel<<<KK, 1024, 0, stream>>>(boxes, scores, thres, cand_val, cand_ord);
    topk_kernel<<<1, 1024, 0, stream>>>(boxes, scores, cand_val, cand_ord, out);
}